// DSHLoss_PartSample_48447231099378
// MI455X (gfx1250) — compile-verified
//
#include <hip/hip_runtime.h>
#include <hip/hip_bf16.h>

typedef __attribute__((ext_vector_type(2))) float v2f;
typedef __attribute__((ext_vector_type(8))) float v8f;

#define NB    200000
#define BB    256
#define BITD  64
#define NC    100
#define SS    30
#define PP    (BB*SS)      /* 7680 pool columns */
#define NTILE (PP/16)      /* 480 */
#define MTILE (BB/16)      /* 16  */
#define MCONST 128.0f      /* m = 2*bit */
#define ALPHA  0.01f

/* ---- workspace layout (bytes) ---- */
#define OFF_YS    0u         /* 200000 f32  scattered Y            */
#define OFF_LIST  800000u    /* 100*30 i32  per-class first-30 idx */
#define OFF_CNT   812000u    /* 100 i32     per-class count (<=30) */
#define OFF_UCLS  812416u    /* 3000*64 f32 pool rows (zero pad)   */
#define OFF_NCLS  1580416u   /* 3000 f32    pool row sq-norms      */
#define OFF_NU    1592448u   /* 256 f32     sample sq-norms        */
#define OFF_L2    1593472u   /* 1 f32       L1-reg sum             */
#define OFF_PART  1593600u   /* 7680 f32    per-tile loss partials */
/* total 1624320 bytes (~1.55 MB) */

__global__ void k_copyY(const float* __restrict__ Y, float* __restrict__ ys) {
    for (int i = blockIdx.x * blockDim.x + threadIdx.x; i < NB; i += gridDim.x * blockDim.x)
        ys[i] = Y[i];
}

__global__ void k_scatterY(const int* __restrict__ y, const int* __restrict__ ind,
                           float* __restrict__ ys) {
    int t = threadIdx.x;                  /* 256 threads */
    ys[ind[t]] = (float)y[t];
}

/* one wave per class: ordered first-30 ascending scan of ys */
__global__ void k_classlist(const float* __restrict__ ys, int* __restrict__ list,
                            int* __restrict__ cnt_out) {
    __shared__ int sflags[32];
    int c = blockIdx.x, lane = threadIdx.x;
    float cf = (float)c;
    int cnt = 0;
    for (int base = 0; base < NB; base += 32) {   /* NB is a multiple of 32 */
        int idx = base + lane;
        int m = (ys[idx] == cf) ? 1 : 0;
        sflags[lane] = m;
        __syncthreads();
        int pre = 0, tot = 0;
        for (int t = 0; t < 32; ++t) { int f = sflags[t]; tot += f; pre += (t < lane) ? f : 0; }
        if (m) { int pos = cnt + pre; if (pos < SS) list[c * SS + pos] = idx; }
        cnt += tot;
        __syncthreads();
        if (cnt >= SS) break;
    }
    if (lane == 0) cnt_out[c] = (cnt < SS) ? cnt : SS;
}

/* one block per (class,slot): gather pool row (scatter-aware) + row norm */
__global__ void k_buildpool(const float* __restrict__ u, const float* __restrict__ U,
                            const int* __restrict__ ind, const int* __restrict__ list,
                            const int* __restrict__ cnt, float* __restrict__ ucls,
                            float* __restrict__ ncls) {
    __shared__ float sred[64];
    int b = blockIdx.x;           /* 0..2999 */
    int c = b / SS, s = b - c * SS;
    int t = threadIdx.x;          /* 64 threads */
    float val = 0.0f;
    if (s < cnt[c]) {
        int j = list[b];
        int owner = -1;
        for (int k = 0; k < BB; ++k) if (ind[k] == j) { owner = k; break; }
        const float* src = (owner >= 0) ? (u + (size_t)owner * BITD)
                                        : (U + (size_t)j * BITD);
        val = src[t];
    }
    ucls[(size_t)b * BITD + t] = val;
    sred[t] = val * val;
    __syncthreads();
    for (int off = 32; off > 0; off >>= 1) {
        if (t < off) sred[t] += sred[t + off];
        __syncthreads();
    }
    if (t == 0) ncls[b] = sred[0];
}

/* sample squared norms + L1 regularizer (single block, deterministic) */
__global__ void k_nu_l2(const float* __restrict__ u, float* __restrict__ nu,
                        float* __restrict__ l2out) {
    __shared__ float sred[BB];
    int i = threadIdx.x;          /* 256 threads */
    float nn = 0.0f, l2 = 0.0f;
    for (int k = 0; k < BITD; ++k) {
        float x = u[(size_t)i * BITD + k];
        nn += x * x;
        l2 += fabsf(fabsf(x) - 1.0f);
    }
    nu[i] = nn;
    sred[i] = l2;
    __syncthreads();
    for (int off = 128; off > 0; off >>= 1) {
        if (i < off) sred[i] += sred[i + off];
        __syncthreads();
    }
    if (i == 0) l2out[0] = sred[0];
}

/* main WMMA kernel: one wave32 per 16x16 tile of the 256x7680 distance matrix */
__global__ void k_gemm_loss(const float* __restrict__ u, const int* __restrict__ y,
                            const float* __restrict__ ucls, const float* __restrict__ ncls,
                            const int* __restrict__ cnt, const float* __restrict__ nu,
                            float* __restrict__ part) {
    int lane = threadIdx.x;               /* 32 */
    int n0 = blockIdx.x * 16;             /* pool column base  (0..479 tiles) */
    int m0 = blockIdx.y * 16;             /* sample row base   (0..15 tiles)  */

    int nloc = lane & 15;                 /* column (and A-row) index in tile */
    int p  = n0 + nloc;                   /* pool column */
    int ip = p / SS, s = p - ip * SS;     /* owning sample, slot */
    int c  = y[ip];                       /* pool column class   */
    int rowid = c * SS + s;
    const float* bp = ucls + (size_t)rowid * BITD;
    float vmaskf = (s < cnt[c]) ? 1.0f : 0.0f;
    float npcol  = ncls[rowid];
    float ycol   = (float)c;

    int kb = (lane >> 4) * 2;             /* K sub-offset: lanes 0-15 -> K0/K1, 16-31 -> K2/K3 */
    const float* ap = u + (size_t)(m0 + nloc) * BITD;

    v8f acc = {};
    for (int k0 = 0; k0 < BITD; k0 += 4) {
        v2f a, b;
        a.x = ap[k0 + kb];  a.y = ap[k0 + kb + 1];
        b.x = bp[k0 + kb];  b.y = bp[k0 + kb + 1];
        acc = __builtin_amdgcn_wmma_f32_16x16x4_f32(
            /*neg_a=*/false, a, /*neg_b=*/false, b,
            /*c_mod=*/(short)0, acc, /*reuse_a=*/false, /*reuse_b=*/false);
    }

    /* C/D layout: lane<16 rows m0+0..7 ; lane>=16 rows m0+8..15 */
    int rbase = m0 + ((lane < 16) ? 0 : 8);
    float sum = 0.0f;
    for (int r = 0; r < 8; ++r) {
        int row = rbase + r;
        float yrow = (float)y[row];
        float dist = nu[row] + npcol - 2.0f * acc[r];
        float neg  = 0.5f * fmaxf(MCONST - dist, 0.0f);
        float pos  = 0.5f * dist;
        float pair = (yrow != ycol) ? neg : pos;
        sum += pair * vmaskf;
    }
    for (int off = 16; off > 0; off >>= 1) sum += __shfl_down(sum, off, 32);
    if (lane == 0) part[blockIdx.y * NTILE + blockIdx.x] = sum;
}

/* final deterministic reduction: loss = sum(part)/(B*step) + alpha*l2/(B*bit) */
__global__ void k_final(const float* __restrict__ part, const float* __restrict__ l2,
                        const int* __restrict__ y, const int* __restrict__ cnt,
                        float* __restrict__ out) {
    __shared__ float sf[BB];
    __shared__ int   si[BB];
    int t = threadIdx.x;                  /* 256 threads */
    float ps = 0.0f;
    for (int k = 0; k < SS; ++k) ps += part[t * SS + k];   /* 7680 = 256*30 */
    int c = y[t];
    int cc = cnt[c];
    sf[t] = ps;
    si[t] = (cc < SS) ? cc : SS;
    __syncthreads();
    for (int off = 128; off > 0; off >>= 1) {
        if (t < off) { sf[t] += sf[t + off]; si[t] += si[t + off]; }
        __syncthreads();
    }
    if (t == 0) {
        float step  = (float)si[0];
        float loss1 = (step > 0.0f) ? sf[0] / ((float)BB * step) : 0.0f;
        float loss2 = ALPHA * l2[0] / ((float)(BB * BITD));
        out[0] = loss1 + loss2;
    }
}

extern "C" void kernel_launch(void* const* d_in, const int* in_sizes, int n_in,
                              void* d_out, int out_size, void* d_ws, size_t ws_size,
                              hipStream_t stream) {
    const float* u   = (const float*)d_in[0];
    const int*   y   = (const int*)d_in[1];
    const int*   ind = (const int*)d_in[2];
    const float* U   = (const float*)d_in[3];
    const float* Y   = (const float*)d_in[4];
    float* out = (float*)d_out;

    char* ws = (char*)d_ws;
    float* ys   = (float*)(ws + OFF_YS);
    int*   list = (int*)  (ws + OFF_LIST);
    int*   cnt  = (int*)  (ws + OFF_CNT);
    float* ucls = (float*)(ws + OFF_UCLS);
    float* ncls = (float*)(ws + OFF_NCLS);
    float* nu   = (float*)(ws + OFF_NU);
    float* l2   = (float*)(ws + OFF_L2);
    float* part = (float*)(ws + OFF_PART);

    k_copyY     <<<782, 256, 0, stream>>>(Y, ys);
    k_scatterY  <<<1, 256, 0, stream>>>(y, ind, ys);
    k_classlist <<<NC, 32, 0, stream>>>(ys, list, cnt);
    k_buildpool <<<NC * SS, 64, 0, stream>>>(u, U, ind, list, cnt, ucls, ncls);
    k_nu_l2     <<<1, BB, 0, stream>>>(u, nu, l2);
    k_gemm_loss <<<dim3(NTILE, MTILE), 32, 0, stream>>>(u, y, ucls, ncls, cnt, nu, part);
    k_final     <<<1, BB, 0, stream>>>(part, l2, y, cnt, out);
}